// NamedSlotMemory_32607391711241
// MI455X (gfx1250) — compile-verified
//
#include <hip/hip_runtime.h>
#include <hip/hip_bf16.h>

// Sizes from the reference
#define BB 32
#define TT 4096
#define DD 256
#define NN 16
#define MLPD 512

typedef __attribute__((ext_vector_type(2))) float v2f;
typedef __attribute__((ext_vector_type(8))) float v8f;

// D = A(16x4) * B(4x16) + C, fp32 WMMA (CDNA5)
#define WMMA4(a, b, c) __builtin_amdgcn_wmma_f32_16x16x4_f32( \
    false, (a), false, (b), (short)0, (c), false, false)

__device__ __forceinline__ float redsum32(float v) {
  #pragma unroll
  for (int m = 16; m >= 1; m >>= 1) v += __shfl_xor(v, m, 32);
  return v;
}
__device__ __forceinline__ float redsum16(float v) {
  #pragma unroll
  for (int m = 8; m >= 1; m >>= 1) v += __shfl_xor(v, m, 32);
  return v;
}
__device__ __forceinline__ float redmax16(float v) {
  #pragma unroll
  for (int m = 8; m >= 1; m >>= 1) v = fmaxf(v, __shfl_xor(v, m, 32));
  return v;
}

// slots[b][n][d] = slot_init[n][d]
__global__ void init_slots_kernel(const float* __restrict__ si, float* __restrict__ slots) {
  int i = blockIdx.x * 256 + threadIdx.x;          // 131072 total
  slots[i] = si[i & (NN * DD - 1)];
}

// LN(x) -> LDS; k = xn@Wk^T+bk; v = xn@Wv^T+bv; also out[:, :256] = x
__global__ __launch_bounds__(1024) void ln_kv_kernel(
    const float* __restrict__ x, const float* __restrict__ lng, const float* __restrict__ lnb,
    const float* __restrict__ Wk, const float* __restrict__ bk,
    const float* __restrict__ Wv, const float* __restrict__ bv,
    float* __restrict__ k, float* __restrict__ v, float* __restrict__ out) {
  __shared__ float xs[16][260];
  const int tid = threadIdx.x;
  const int wave = tid >> 5, lane = tid & 31;
  const long rb = (long)blockIdx.x * 16;

  if (wave < 16) {  // LayerNorm: one wave per row
    const float* xr = x + (rb + wave) * DD;
    float vals[8], s = 0.f, sq = 0.f;
    #pragma unroll
    for (int j = 0; j < 8; ++j) { float t = xr[lane * 8 + j]; vals[j] = t; s += t; sq += t * t; }
    s = redsum32(s); sq = redsum32(sq);
    float mean = s * (1.f / 256.f);
    float rstd = rsqrtf(sq * (1.f / 256.f) - mean * mean + 1e-5f);
    float* orow = out + (rb + wave) * 512;
    #pragma unroll
    for (int j = 0; j < 8; ++j) {
      int c = lane * 8 + j;
      xs[wave][c] = (vals[j] - mean) * rstd * lng[c] + lnb[c];
      orow[c] = vals[j];
    }
  }
  __syncthreads();

  const int mat = wave >> 4;          // 0 -> k, 1 -> v
  const int n0 = (wave & 15) << 4;
  const float* W = mat ? Wv : Wk;
  const float* bias = mat ? bv : bk;
  float* dst = mat ? v : k;
  const int hx = lane >> 4, lr = lane & 15;
  const float* arow = &xs[lr][2 * hx];
  const float* wrow = W + (n0 + lr) * DD + 2 * hx;
  v8f acc = {0.f, 0.f, 0.f, 0.f, 0.f, 0.f, 0.f, 0.f};
  #pragma unroll 8
  for (int kk = 0; kk < DD; kk += 4) {
    v2f a; a.x = arow[kk]; a.y = arow[kk + 1];
    v2f b = *(const v2f*)(wrow + kk);
    acc = WMMA4(a, b, acc);
  }
  float bv_ = bias[n0 + lr];
  #pragma unroll
  for (int r = 0; r < 8; ++r)
    dst[(rb + r + 8 * hx) * DD + n0 + lr] = acc[r] + bv_;
}

// q = LN(slots)@Wq^T + bq ; zero colsum, updates
__global__ __launch_bounds__(512) void slots_q_kernel(
    const float* __restrict__ slots, const float* __restrict__ lng, const float* __restrict__ lnb,
    const float* __restrict__ Wq, const float* __restrict__ bq,
    float* __restrict__ q, float* __restrict__ colsum, float* __restrict__ updates) {
  __shared__ float ls[16][260];
  const int b = blockIdx.x, tid = threadIdx.x;
  const int wave = tid >> 5, lane = tid & 31;
  if (tid < 16) colsum[b * 16 + tid] = 0.f;
  #pragma unroll
  for (int j = 0; j < 8; ++j) updates[b * 4096 + j * 512 + tid] = 0.f;
  {
    const float* sr = slots + (b * 16 + wave) * DD;
    float vals[8], s = 0.f, sq = 0.f;
    #pragma unroll
    for (int j = 0; j < 8; ++j) { float t = sr[lane * 8 + j]; vals[j] = t; s += t; sq += t * t; }
    s = redsum32(s); sq = redsum32(sq);
    float mean = s * (1.f / 256.f);
    float rstd = rsqrtf(sq * (1.f / 256.f) - mean * mean + 1e-5f);
    #pragma unroll
    for (int j = 0; j < 8; ++j) {
      int c = lane * 8 + j;
      ls[wave][c] = (vals[j] - mean) * rstd * lng[c] + lnb[c];
    }
  }
  __syncthreads();
  const int hx = lane >> 4, lr = lane & 15;
  const int n0 = wave << 4;
  const float* arow = &ls[lr][2 * hx];
  const float* wrow = Wq + (n0 + lr) * DD + 2 * hx;
  v8f acc = {0.f, 0.f, 0.f, 0.f, 0.f, 0.f, 0.f, 0.f};
  #pragma unroll 8
  for (int kk = 0; kk < DD; kk += 4) {
    v2f a; a.x = arow[kk]; a.y = arow[kk + 1];
    v2f b = *(const v2f*)(wrow + kk);
    acc = WMMA4(a, b, acc);
  }
  float bv_ = bq[n0 + lr];
  #pragma unroll
  for (int r = 0; r < 8; ++r)
    q[(b * 16 + r + 8 * hx) * DD + n0 + lr] = acc[r] + bv_;
}

// logits = k@q^T * scale (WMMA); softmax over N; attn out; colsum += attn
__global__ __launch_bounds__(512) void attn_kernel(
    const float* __restrict__ k, const float* __restrict__ q,
    float* __restrict__ attn, float* __restrict__ colsum) {
  __shared__ float qs[16][260];
  __shared__ float cs[16];
  const int b = blockIdx.y, tid = threadIdx.x;
  const int wave = tid >> 5, lane = tid & 31;
  #pragma unroll
  for (int j = 0; j < 8; ++j) {
    int e = j * 512 + tid;
    qs[e >> 8][e & 255] = q[b * 4096 + e];
  }
  if (tid < 16) cs[tid] = 0.f;
  __syncthreads();
  const int hx = lane >> 4, lr = lane & 15;
  const long t0 = (long)b * TT + blockIdx.x * 256 + wave * 16;
  const float* krow = k + (t0 + lr) * DD + 2 * hx;
  const float* qrow = &qs[lr][2 * hx];
  v8f acc = {0.f, 0.f, 0.f, 0.f, 0.f, 0.f, 0.f, 0.f};
  #pragma unroll 8
  for (int kk = 0; kk < DD; kk += 4) {
    v2f a = *(const v2f*)(krow + kk);
    v2f bvec; bvec.x = qrow[kk]; bvec.y = qrow[kk + 1];
    acc = WMMA4(a, bvec, acc);
  }
  const float scale = 0.0625f;  // 1/sqrt(256)
  float part = 0.f;
  #pragma unroll
  for (int r = 0; r < 8; ++r) {
    float l = acc[r] * scale;
    float m = redmax16(l);
    float e = __expf(l - m);
    float s = redsum16(e);
    float a = e / s;
    attn[(t0 + r + 8 * hx) * NN + lr] = a;
    part += a;
  }
  atomicAdd(&cs[lr], part);
  __syncthreads();
  if (tid < 16) atomicAdd(&colsum[b * 16 + tid], cs[tid]);
}

// updates[b,n,:] += sum_t (attn/colsum)[t,n] * v[t,:]  (WMMA, A = weights^T)
__global__ __launch_bounds__(512) void updates_kernel(
    const float* __restrict__ v, const float* __restrict__ attn,
    const float* __restrict__ colsum, float* __restrict__ updates) {
  __shared__ float wgt[256][16];
  __shared__ float cs[16];
  const int b = blockIdx.y, tid = threadIdx.x;
  if (tid < 16) cs[tid] = colsum[b * 16 + tid] + 1e-8f;
  __syncthreads();
  const long t0 = (long)b * TT + blockIdx.x * 256;
  #pragma unroll
  for (int j = 0; j < 8; ++j) {
    int e = j * 512 + tid;
    wgt[e >> 4][e & 15] = attn[t0 * NN + e] / cs[e & 15];
  }
  __syncthreads();
  const int wave = tid >> 5, lane = tid & 31, hx = lane >> 4, lr = lane & 15;
  const int d0 = wave << 4;
  const float* vb = v + (t0 + 2 * hx) * DD + d0 + lr;
  v8f acc = {0.f, 0.f, 0.f, 0.f, 0.f, 0.f, 0.f, 0.f};
  #pragma unroll 4
  for (int t = 0; t < 256; t += 4) {
    v2f a; a.x = wgt[t + 2 * hx][lr]; a.y = wgt[t + 2 * hx + 1][lr];
    v2f bvec; bvec.x = vb[t * DD]; bvec.y = vb[(t + 1) * DD];
    acc = WMMA4(a, bvec, acc);
  }
  #pragma unroll
  for (int r = 0; r < 8; ++r)
    atomicAdd(&updates[(b * 16 + r + 8 * hx) * DD + d0 + lr], acc[r]);
}

// gx = updates@Wih^T + bih, gh = slots@Whh^T + bhh -> gbuf[b*16+row][0:768 | 768:1536]
__global__ __launch_bounds__(512) void gru_gemm_kernel(
    const float* __restrict__ updates, const float* __restrict__ slots,
    const float* __restrict__ Wih, const float* __restrict__ bih,
    const float* __restrict__ Whh, const float* __restrict__ bhh,
    float* __restrict__ gbuf) {
  __shared__ float us[16][260];
  __shared__ float sh[16][260];
  const int b = blockIdx.x, tid = threadIdx.x;
  #pragma unroll
  for (int j = 0; j < 8; ++j) {
    int e = j * 512 + tid;
    us[e >> 8][e & 255] = updates[b * 4096 + e];
    sh[e >> 8][e & 255] = slots[b * 4096 + e];
  }
  __syncthreads();
  const int wave = tid >> 5, lane = tid & 31, hx = lane >> 4, lr = lane & 15;
  for (int i = 0; i < 6; ++i) {
    int t = wave * 6 + i;               // 96 tiles: 48 gx + 48 gh
    int isx = (t < 48);
    int n0 = (isx ? t : t - 48) << 4;
    const float* W = isx ? Wih : Whh;
    const float* bias = isx ? bih : bhh;
    const float* arow = isx ? &us[lr][2 * hx] : &sh[lr][2 * hx];
    const float* wrow = W + (n0 + lr) * DD + 2 * hx;
    v8f acc = {0.f, 0.f, 0.f, 0.f, 0.f, 0.f, 0.f, 0.f};
    #pragma unroll 8
    for (int kk = 0; kk < DD; kk += 4) {
      v2f a; a.x = arow[kk]; a.y = arow[kk + 1];
      v2f bvec = *(const v2f*)(wrow + kk);
      acc = WMMA4(a, bvec, acc);
    }
    int dcol = (isx ? 0 : 768) + n0 + lr;
    float bb = bias[n0 + lr];
    #pragma unroll
    for (int r = 0; r < 8; ++r)
      gbuf[((long)b * 16 + r + 8 * hx) * 1536 + dcol] = acc[r] + bb;
  }
}

// gates -> s_new (in slots); LN; hid=relu(h@W1^T+b1); slots = s_new + hid@W2^T + b2
__global__ __launch_bounds__(512) void gru_mlp_kernel(
    const float* __restrict__ gbuf, float* __restrict__ slots,
    const float* __restrict__ lng, const float* __restrict__ lnb,
    const float* __restrict__ W1, const float* __restrict__ b1,
    const float* __restrict__ W2, const float* __restrict__ b2) {
  __shared__ float h[16][260];
  __shared__ float hid[16][516];
  const int b = blockIdx.x, tid = threadIdx.x;
  const int wave = tid >> 5, lane = tid & 31, hx = lane >> 4, lr = lane & 15;
  #pragma unroll
  for (int j = 0; j < 8; ++j) {
    int e = j * 512 + tid;
    int row = e >> 8, c = e & 255;
    const float* gr = gbuf + ((long)b * 16 + row) * 1536;
    float rx = gr[c], zx = gr[256 + c], nx = gr[512 + c];
    float rh = gr[768 + c], zh = gr[1024 + c], nh = gr[1280 + c];
    float r = 1.f / (1.f + __expf(-(rx + rh)));
    float z = 1.f / (1.f + __expf(-(zx + zh)));
    float n = tanhf(nx + r * nh);
    long si = (long)b * 4096 + e;
    float so = slots[si];
    slots[si] = (1.f - z) * n + z * so;   // s_new
  }
  __threadfence_block();
  __syncthreads();
  {
    const float* sr = slots + (long)b * 4096 + wave * 256;
    float vals[8], s = 0.f, sq = 0.f;
    #pragma unroll
    for (int j = 0; j < 8; ++j) { float t = sr[lane * 8 + j]; vals[j] = t; s += t; sq += t * t; }
    s = redsum32(s); sq = redsum32(sq);
    float mean = s * (1.f / 256.f);
    float rstd = rsqrtf(sq * (1.f / 256.f) - mean * mean + 1e-5f);
    #pragma unroll
    for (int j = 0; j < 8; ++j) {
      int c = lane * 8 + j;
      h[wave][c] = (vals[j] - mean) * rstd * lng[c] + lnb[c];
    }
  }
  __syncthreads();
  for (int i = 0; i < 2; ++i) {       // MLP1: 32 n-tiles over 16 waves
    int n0 = (wave * 2 + i) << 4;
    const float* arow = &h[lr][2 * hx];
    const float* wrow = W1 + (n0 + lr) * DD + 2 * hx;
    v8f acc = {0.f, 0.f, 0.f, 0.f, 0.f, 0.f, 0.f, 0.f};
    #pragma unroll 8
    for (int kk = 0; kk < DD; kk += 4) {
      v2f a; a.x = arow[kk]; a.y = arow[kk + 1];
      v2f bvec = *(const v2f*)(wrow + kk);
      acc = WMMA4(a, bvec, acc);
    }
    float bb = b1[n0 + lr];
    #pragma unroll
    for (int r = 0; r < 8; ++r)
      hid[r + 8 * hx][n0 + lr] = fmaxf(acc[r] + bb, 0.f);
  }
  __syncthreads();
  {                                    // MLP2: K=512
    int n0 = wave << 4;
    const float* arow = &hid[lr][2 * hx];
    const float* wrow = W2 + (n0 + lr) * MLPD + 2 * hx;
    v8f acc = {0.f, 0.f, 0.f, 0.f, 0.f, 0.f, 0.f, 0.f};
    #pragma unroll 8
    for (int kk = 0; kk < MLPD; kk += 4) {
      v2f a; a.x = arow[kk]; a.y = arow[kk + 1];
      v2f bvec = *(const v2f*)(wrow + kk);
      acc = WMMA4(a, bvec, acc);
    }
    float bb = b2[n0 + lr];
    #pragma unroll
    for (int r = 0; r < 8; ++r) {
      long si = ((long)b * 16 + r + 8 * hx) * DD + n0 + lr;
      slots[si] += acc[r] + bb;
    }
  }
}

// out[:, 256:512] = attn @ slots (bandwidth bound; slots in registers)
__global__ __launch_bounds__(256) void output_kernel(
    const float* __restrict__ attn, const float* __restrict__ slots,
    float* __restrict__ out) {
  __shared__ float at[64 * 16];
  const int b = blockIdx.y, tid = threadIdx.x;
  const long t0 = (long)b * TT + blockIdx.x * 64;
  #pragma unroll
  for (int j = 0; j < 4; ++j) at[j * 256 + tid] = attn[t0 * NN + j * 256 + tid];
  float sl[16];
  #pragma unroll
  for (int n = 0; n < 16; ++n) sl[n] = slots[(b * 16 + n) * DD + tid];
  __syncthreads();
  for (int t = 0; t < 64; ++t) {
    float acc = 0.f;
    #pragma unroll
    for (int n = 0; n < 16; ++n) acc += at[t * 16 + n] * sl[n];
    out[(t0 + t) * 512 + 256 + tid] = acc;
  }
}

extern "C" void kernel_launch(void* const* d_in, const int* in_sizes, int n_in,
                              void* d_out, int out_size, void* d_ws, size_t ws_size,
                              hipStream_t stream) {
  const float* x      = (const float*)d_in[0];
  const float* ln_in_g = (const float*)d_in[1];
  const float* ln_in_b = (const float*)d_in[2];
  const float* ln_sg  = (const float*)d_in[3];
  const float* ln_sb  = (const float*)d_in[4];
  const float* ln_mg  = (const float*)d_in[5];
  const float* ln_mb  = (const float*)d_in[6];
  const float* Wq = (const float*)d_in[7];
  const float* bq = (const float*)d_in[8];
  const float* Wk = (const float*)d_in[9];
  const float* bk = (const float*)d_in[10];
  const float* Wv = (const float*)d_in[11];
  const float* bv = (const float*)d_in[12];
  const float* Wih = (const float*)d_in[13];
  const float* bih = (const float*)d_in[14];
  const float* Whh = (const float*)d_in[15];
  const float* bhh = (const float*)d_in[16];
  const float* W1 = (const float*)d_in[17];
  const float* b1 = (const float*)d_in[18];
  const float* W2 = (const float*)d_in[19];
  const float* b2 = (const float*)d_in[20];
  const float* slot_init = (const float*)d_in[21];

  float* out = (float*)d_out;
  float* ws = (float*)d_ws;
  // workspace layout (floats)
  float* k       = ws;                         // 33,554,432
  float* v       = ws + 33554432;              // 33,554,432
  float* attn    = ws + 67108864;              // 2,097,152
  float* q       = ws + 69206016;              // 131,072
  float* colsum  = ws + 69337088;              // 512
  float* updates = ws + 69337600;              // 131,072
  float* gbuf    = ws + 69468672;              // 786,432
  float* slots   = out + (long)BB * TT * 512;  // final slots output in place

  init_slots_kernel<<<512, 256, 0, stream>>>(slot_init, slots);
  ln_kv_kernel<<<BB * TT / 16, 1024, 0, stream>>>(x, ln_in_g, ln_in_b, Wk, bk, Wv, bv, k, v, out);
  for (int it = 0; it < 3; ++it) {
    slots_q_kernel<<<BB, 512, 0, stream>>>(slots, ln_sg, ln_sb, Wq, bq, q, colsum, updates);
    attn_kernel<<<dim3(TT / 256, BB), 512, 0, stream>>>(k, q, attn, colsum);
    updates_kernel<<<dim3(TT / 256, BB), 512, 0, stream>>>(v, attn, colsum, updates);
    gru_gemm_kernel<<<BB, 512, 0, stream>>>(updates, slots, Wih, bih, Whh, bhh, gbuf);
    gru_mlp_kernel<<<BB, 512, 0, stream>>>(gbuf, slots, ln_mg, ln_mb, W1, b1, W2, b2);
  }
  output_kernel<<<dim3(TT / 64, BB), 256, 0, stream>>>(attn, slots, out);
}